// MultitaskLossNoRedundancy_81999515615829
// MI455X (gfx1250) — compile-verified
//
#include <hip/hip_runtime.h>
#include <math.h>

typedef float v2f __attribute__((ext_vector_type(2)));
typedef float v4f __attribute__((ext_vector_type(4)));
typedef float v8f __attribute__((ext_vector_type(8)));

__device__ __forceinline__ float softplus_stable(float x) {
    // softplus(x) = max(x,0) + log1p(exp(-|x|))  (numerically stable)
    return fmaxf(x, 0.0f) + log1pf(__expf(-fabsf(x)));
}

// Kernel 1: streaming pass over all B rows. Each thread accumulates a float
// partial; waves reduce via shuffles, block reduces via LDS, one partial per
// block written to d_ws. Fully deterministic.
__global__ void __launch_bounds__(256)
multitask_loss_partial(const float* __restrict__ pb,   // [B]   pred binary logits
                       const v4f*   __restrict__ pt,   // [B,4] pred type logits
                       const v4f*   __restrict__ ps,   // [B,16] pred source logits (4x v4f per row)
                       const float* __restrict__ tb,   // [B]   true binary
                       const v4f*   __restrict__ tt,   // [B,4] true type
                       const int*   __restrict__ lab,  // [B]   true source class
                       float* __restrict__ block_out,  // [gridDim.x]
                       int B)
{
    const int tid    = blockIdx.x * blockDim.x + threadIdx.x;
    const int stride = gridDim.x * blockDim.x;

    float acc = 0.0f;

    for (int i = tid; i < B; i += stride) {
        // ---- binary BCE:  y*sp(-x)+(1-y)*sp(x) == sp(x) - x*y ----
        float x = __builtin_nontemporal_load(pb + i);
        float y = __builtin_nontemporal_load(tb + i);
        float l = softplus_stable(x) - x * y;
        bool binary_c = (x >= 0.0f);                       // sigmoid(x) >= 0.5

        // ---- type BCE (mean over B*T => 0.25 * row-sum here) ----
        v4f xt = __builtin_nontemporal_load(&pt[i]);
        v4f yt = __builtin_nontemporal_load(&tt[i]);
        float bt = 0.0f;
        bool type_c = false;
#pragma unroll
        for (int j = 0; j < 4; ++j) {
            bt += softplus_stable(xt[j]) - xt[j] * yt[j];
            type_c |= (xt[j] >= 0.0f);
        }
        l += 0.25f * bt;

        // ---- 16-way cross entropy: lse(ps) - ps[label] ----
        v4f s[4];
#pragma unroll
        for (int v = 0; v < 4; ++v)
            s[v] = __builtin_nontemporal_load(&ps[(size_t)4 * i + v]);

        float mx = s[0][0];
#pragma unroll
        for (int v = 0; v < 4; ++v)
#pragma unroll
            for (int j = 0; j < 4; ++j)
                mx = fmaxf(mx, s[v][j]);

        float es = 0.0f;
#pragma unroll
        for (int v = 0; v < 4; ++v)
#pragma unroll
            for (int j = 0; j < 4; ++j)
                es += __expf(s[v][j] - mx);

        int li = __builtin_nontemporal_load(lab + i);
        float plab = s[0][0];
#pragma unroll
        for (int v = 0; v < 4; ++v)
#pragma unroll
            for (int j = 0; j < 4; ++j)
                if (v * 4 + j > 0)
                    plab = (li == v * 4 + j) ? s[v][j] : plab;

        l += (mx + __logf(es)) - plab;

        // ---- consistency XOR terms (ALPHA = 1) ----
        // argmax(ps) > 0  <=>  max(ps[1:]) > ps[0]   (first-max tie rule)
        float mrest = s[0][1];
#pragma unroll
        for (int v = 0; v < 4; ++v)
#pragma unroll
            for (int j = 0; j < 4; ++j)
                if (v * 4 + j > 1)
                    mrest = fmaxf(mrest, s[v][j]);
        bool source_c = (mrest > s[0][0]);

        l += (binary_c != type_c)   ? 1.0f : 0.0f;
        l += (binary_c != source_c) ? 1.0f : 0.0f;

        acc += l;
    }

    // wave32 reduction
#pragma unroll
    for (int m = 16; m >= 1; m >>= 1)
        acc += __shfl_xor(acc, m, 32);

    __shared__ float sm[8];
    const int lane = threadIdx.x & 31;
    const int wid  = threadIdx.x >> 5;
    if (lane == 0) sm[wid] = acc;
    __syncthreads();
    if (threadIdx.x == 0) {
        float sblk = 0.0f;
        const int nw = blockDim.x >> 5;
        for (int w = 0; w < nw; ++w) sblk += sm[w];   // fixed order: deterministic
        block_out[blockIdx.x] = sblk;
    }
}

// Kernel 2: one wave. Sums block partials in fixed per-lane order, then uses
// V_WMMA_F32_16X16X4_F32 with an all-ones A-matrix: D = 1 x B gives column
// sums of B (lane n and n+16 occupy distinct K rows of column n, so each
// column sum = p[n] + p[n+16] regardless of exact K assignment). A 4-step
// shuffle-xor over the 16 columns finishes the reduction.
__global__ void __launch_bounds__(32)
multitask_loss_finalize(const float* __restrict__ partials, int n,
                        float* __restrict__ out, float scale)
{
    const int lane = threadIdx.x;   // 0..31
    float p = 0.0f;
    for (int k = lane; k < n; k += 32)
        p += partials[k];

    v2f a; a.x = 1.0f; a.y = 1.0f;      // A = ones (16x4 f32)
    v2f b; b.x = p;    b.y = 0.0f;      // B holds one partial per lane
    v8f c = {};
    // 8-arg form: (neg_a, A, neg_b, B, c_mod, C, reuse_a, reuse_b)
    v8f d = __builtin_amdgcn_wmma_f32_16x16x4_f32(
        false, a, false, b, (short)0, c, false, false);

    float cs = d[0];                    // = colsum_{lane % 16}
#pragma unroll
    for (int m = 8; m >= 1; m >>= 1)    // sum over the 16 distinct columns
        cs += __shfl_xor(cs, m, 32);

    if (lane == 0) out[0] = cs * scale;
}

extern "C" void kernel_launch(void* const* d_in, const int* in_sizes, int n_in,
                              void* d_out, int out_size, void* d_ws, size_t ws_size,
                              hipStream_t stream) {
    const float* pb  = (const float*)d_in[0];
    const v4f*   pt  = (const v4f*)  d_in[1];
    const v4f*   ps  = (const v4f*)  d_in[2];
    const float* tb  = (const float*)d_in[3];
    const v4f*   tt  = (const v4f*)  d_in[4];
    const int*   lab = (const int*)  d_in[5];

    const int B = in_sizes[0];

    const int threads = 256;
    const int elems_per_thread = 4;
    int blocks = (B + threads * elems_per_thread - 1) / (threads * elems_per_thread);
    if (blocks > 4096) blocks = 4096;
    if (blocks < 1)    blocks = 1;
    // d_ws scratch: one float per block (<= 16 KB)
    if ((size_t)blocks * sizeof(float) > ws_size)
        blocks = (int)(ws_size / sizeof(float));

    float* partials = (float*)d_ws;

    multitask_loss_partial<<<blocks, threads, 0, stream>>>(
        pb, pt, ps, tb, tt, lab, partials, B);

    multitask_loss_finalize<<<1, 32, 0, stream>>>(
        partials, blocks, (float*)d_out, 1.0f / (float)B);
}